// Full_Graph_51539607552373
// MI455X (gfx1250) — compile-verified
//
#include <hip/hip_runtime.h>

typedef __attribute__((ext_vector_type(2))) float v2f;
typedef __attribute__((ext_vector_type(8))) float v8f;

#define HW 262144      // 512*512
#define NB 8
#define CH 10
#define TOT_TILES (NB * (HW / 16))

// ---- helpers -------------------------------------------------------------

// Single-instruction ReLU: v_med3_f32(x, 0, +inf) == clamp below zero.
// (NaN input -> med3 returns min of remaining = 0, matching max(x,0).)
__device__ __forceinline__ float relu(float x) {
  return __builtin_amdgcn_fmed3f(x, 0.f, __builtin_inff());
}

// swap lane halves (lane L <-> lane L^16) via ds_swizzle SWAPX16:
// group-of-32 mode, xor=0x10, or=0, and=0x1f  -> offset 0x401f
__device__ __forceinline__ float swap16(float x) {
  int i = __builtin_bit_cast(int, x);
  i = __builtin_amdgcn_ds_swizzle(i, 0x401f);
  return __builtin_bit_cast(float, i);
}

// D = A(16x4 f32) * B(4x16 f32) + C(16x16 f32), full fp32 precision
__device__ __forceinline__ v8f wmma4(v2f a, v2f b, v8f c) {
  return __builtin_amdgcn_wmma_f32_16x16x4_f32(
      /*neg_a=*/false, a, /*neg_b=*/false, b,
      /*c_mod=*/(short)0, c, /*reuse_a=*/false, /*reuse_b=*/false);
}

// Fetch channel `c` (0..31) of an accumulator pair laid out as
// D[t] VGPR v: lanes0-15 = chan 16t+v, lanes16-31 = chan 16t+v+8,
// positioned into the requested lane half (wantHi). c/wantHi are
// compile-time constants after unrolling, so the swap16 branch folds.
__device__ __forceinline__ float fetchHalf(const v8f* d, int c, int wantHi) {
  int r = c & 15;
  float x = d[c >> 4][r & 7];
  int srcHi = (r >> 3) & 1;
  return (srcHi != wantHi) ? swap16(x) : x;
}

// ---- BN-fold prep kernel (weights*scale, bias=shift), padded to 16 rows ---
// ws layout (floats): W1[32][20]@0  W2p[16][32]@640  W3[32][20]@1152
//                     W4p[16][32]@1792  B1[32]@2304 B2p[16]@2336
//                     B3[32]@2352 B4p[16]@2384    (total 2400 floats)
__global__ void fold_kernel(
    const float* w1, const float* g1, const float* b1, const float* m1, const float* v1,
    const float* w2, const float* g2, const float* b2, const float* m2, const float* v2,
    const float* w3, const float* g3, const float* b3, const float* m3, const float* v3,
    const float* w4, const float* g4, const float* b4, const float* m4, const float* v4,
    float* ws) {
  const float eps = 1e-5f;
  int i = blockIdx.x * blockDim.x + threadIdx.x;
  if (i < 640) { int o = i / 20; float s = g1[o] * rsqrtf(v1[o] + eps); ws[i] = w1[i] * s; }
  if (i < 512) {
    int o = i / 32, c = i % 32; float val = 0.f;
    if (o < CH) { float s = g2[o] * rsqrtf(v2[o] + eps); val = w2[o * 32 + c] * s; }
    ws[640 + i] = val;
  }
  if (i < 640) { int o = i / 20; float s = g3[o] * rsqrtf(v3[o] + eps); ws[1152 + i] = w3[i] * s; }
  if (i < 512) {
    int o = i / 32, c = i % 32; float val = 0.f;
    if (o < CH) { float s = g4[o] * rsqrtf(v4[o] + eps); val = w4[o * 32 + c] * s; }
    ws[1792 + i] = val;
  }
  if (i < 32) { float s = g1[i] * rsqrtf(v1[i] + eps); ws[2304 + i] = b1[i] - m1[i] * s; }
  if (i < 16) {
    float val = 0.f;
    if (i < CH) { float s = g2[i] * rsqrtf(v2[i] + eps); val = b2[i] - m2[i] * s; }
    ws[2336 + i] = val;
  }
  if (i < 32) { float s = g3[i] * rsqrtf(v3[i] + eps); ws[2352 + i] = b3[i] - m3[i] * s; }
  if (i < 16) {
    float val = 0.f;
    if (i < CH) { float s = g4[i] * rsqrtf(v4[i] + eps); val = b4[i] - m4[i] * s; }
    ws[2384 + i] = val;
  }
}

// ---- fully fused main kernel: one 16-pixel N-tile per wave iteration ------
__global__ __launch_bounds__(256) void fused_kernel(
    const float* __restrict__ xf, const float* __restrict__ xh1,
    const float* __restrict__ xh2, const float* __restrict__ ws,
    float* __restrict__ out) {
  const int lane = threadIdx.x & 31;
  const int hi = lane >> 4;
  const int ll = lane & 15;

  const float* W1 = ws;
  const float* W2 = ws + 640;
  const float* W3 = ws + 1152;
  const float* W4 = ws + 1792;
  const float* B1 = ws + 2304;
  const float* B2 = ws + 2336;
  const float* B3 = ws + 2352;
  const float* B4 = ws + 2384;

  // Hoist folded weights into A-operand registers.
  // A 16x4 fp32 layout: lane L(0-15): (M=L,K=4j),(M=L,K=4j+1); lane L+16: K=4j+2,4j+3
  v2f a1[2][5], a3[2][5];
#pragma unroll
  for (int mt = 0; mt < 2; ++mt)
#pragma unroll
    for (int j = 0; j < 5; ++j) {
      a1[mt][j] = *(const v2f*)(W1 + (mt * 16 + ll) * 20 + 4 * j + 2 * hi);
      a3[mt][j] = *(const v2f*)(W3 + (mt * 16 + ll) * 20 + 4 * j + 2 * hi);
    }
  v2f a2[8], a4[8];
#pragma unroll
  for (int j = 0; j < 8; ++j) {
    a2[j] = *(const v2f*)(W2 + ll * 32 + 4 * j + 2 * hi);
    a4[j] = *(const v2f*)(W4 + ll * 32 + 4 * j + 2 * hi);
  }
  // Bias seeds for the C accumulator: VGPR v = bias[chan v + 8*hi]
  v8f c1[2], c3[2], c2, c4;
#pragma unroll
  for (int v = 0; v < 8; ++v) {
    c1[0][v] = B1[v + 8 * hi];
    c1[1][v] = B1[16 + v + 8 * hi];
    c3[0][v] = B3[v + 8 * hi];
    c3[1][v] = B3[16 + v + 8 * hi];
    c2[v] = B2[v + 8 * hi];
    c4[v] = B4[v + 8 * hi];
  }

  float* oxf = out;
  float* omsg = out + (size_t)NB * CH * HW;

  const int wavesPerBlock = blockDim.x >> 5;
  int gw = blockIdx.x * wavesPerBlock + (threadIdx.x >> 5);
  int nw = gridDim.x * wavesPerBlock;

  for (int t = gw; t < TOT_TILES; t += nw) {
    int b = t / (HW / 16);
    int hw0 = (t - b * (HW / 16)) * 16;
    int pidx = hw0 + ll;
    const float* xh1b = xh1 + (size_t)b * CH * HW;
    const float* xh2b = xh2 + (size_t)b * CH * HW;
    const float* xfb = xf + (size_t)b * CH * HW;

    // ===== Block 1, conv1: [32x20] x [20x16] (K=20 -> 5 chunks) =====
    v8f d1a = c1[0], d1b = c1[1];
#pragma unroll
    for (int j = 0; j < 5; ++j) {
      int cx = 4 * j + 2 * hi;  // B chunk: .x = chan 4j+2hi, .y = chan +1
      int cy = cx + 1;
      const float* px = (cx < CH) ? (xh1b + (size_t)cx * HW) : (xh2b + (size_t)(cx - CH) * HW);
      const float* py = (cy < CH) ? (xh1b + (size_t)cy * HW) : (xh2b + (size_t)(cy - CH) * HW);
      v2f bb;
      bb.x = px[pidx];
      bb.y = py[pidx];
      d1a = wmma4(a1[0][j], bb, d1a);
      d1b = wmma4(a1[1][j], bb, d1b);
    }
    v8f d1[2];
#pragma unroll
    for (int k = 0; k < 8; ++k) {
      d1[0][k] = relu(d1a[k]);
      d1[1][k] = relu(d1b[k]);
    }

    // ===== Block 1, conv2: [16x32] x [32x16]; B built in-register from d1 ==
    v8f d2 = c2;
#pragma unroll
    for (int j = 0; j < 8; ++j) {
      v2f bb;
      {
        float lov = fetchHalf(d1, 4 * j, 0);
        float hiv = fetchHalf(d1, 4 * j + 2, 1);
        bb.x = hi ? hiv : lov;
      }
      {
        float lov = fetchHalf(d1, 4 * j + 1, 0);
        float hiv = fetchHalf(d1, 4 * j + 3, 1);
        bb.y = hi ? hiv : lov;
      }
      d2 = wmma4(a2[j], bb, d2);
    }
#pragma unroll
    for (int k = 0; k < 8; ++k) d2[k] = relu(d2[k]);

    // store message (chans 0..9 valid)
    float* omsgb = omsg + (size_t)b * CH * HW;
#pragma unroll
    for (int v = 0; v < 8; ++v) {
      int c = v + 8 * hi;
      if (c < CH) omsgb[(size_t)c * HW + pidx] = d2[v];
    }

    // ===== Block 2, conv1: input chans 0-9 = xf, 10-19 = message (in d2) ===
    v8f d3a = c3[0], d3b = c3[1];
#pragma unroll
    for (int j = 0; j < 2; ++j) {  // chans 0..7: pure xf
      int cx = 4 * j + 2 * hi;
      v2f bb;
      bb.x = xfb[(size_t)cx * HW + pidx];
      bb.y = xfb[(size_t)(cx + 1) * HW + pidx];
      d3a = wmma4(a3[0][j], bb, d3a);
      d3b = wmma4(a3[1][j], bb, d3b);
    }
    {  // j=2: lo chans 8,9 from xf; hi chans 10,11 = msg0,msg1 (lo->hi swap)
      float x8 = xfb[(size_t)8 * HW + pidx];
      float x9 = xfb[(size_t)9 * HW + pidx];
      float m0 = swap16(d2[0]);
      float m1 = swap16(d2[1]);
      v2f bb;
      bb.x = hi ? m0 : x8;
      bb.y = hi ? m1 : x9;
      d3a = wmma4(a3[0][2], bb, d3a);
      d3b = wmma4(a3[1][2], bb, d3b);
    }
    {  // j=3: lo chans 12,13 = msg2,3 (already lo); hi chans 14,15 = msg4,5
      float m4 = swap16(d2[4]);
      float m5 = swap16(d2[5]);
      v2f bb;
      bb.x = hi ? m4 : d2[2];
      bb.y = hi ? m5 : d2[3];
      d3a = wmma4(a3[0][3], bb, d3a);
      d3b = wmma4(a3[1][3], bb, d3b);
    }
    {  // j=4: lo chans 16,17 = msg6,7 (lo); hi chans 18,19 = msg8,9 (already hi)
      v2f bb;
      bb.x = hi ? d2[0] : d2[6];
      bb.y = hi ? d2[1] : d2[7];
      d3a = wmma4(a3[0][4], bb, d3a);
      d3b = wmma4(a3[1][4], bb, d3b);
    }
    v8f d3[2];
#pragma unroll
    for (int k = 0; k < 8; ++k) {
      d3[0][k] = relu(d3a[k]);
      d3[1][k] = relu(d3b[k]);
    }

    // ===== Block 2, conv2 =====
    v8f d4 = c4;
#pragma unroll
    for (int j = 0; j < 8; ++j) {
      v2f bb;
      {
        float lov = fetchHalf(d3, 4 * j, 0);
        float hiv = fetchHalf(d3, 4 * j + 2, 1);
        bb.x = hi ? hiv : lov;
      }
      {
        float lov = fetchHalf(d3, 4 * j + 1, 0);
        float hiv = fetchHalf(d3, 4 * j + 3, 1);
        bb.y = hi ? hiv : lov;
      }
      d4 = wmma4(a4[j], bb, d4);
    }
    float* oxfb = oxf + (size_t)b * CH * HW;
#pragma unroll
    for (int v = 0; v < 8; ++v) {
      int c = v + 8 * hi;
      if (c < CH) oxfb[(size_t)c * HW + pidx] = relu(d4[v]);
    }
  }
}

// ---- host-side launcher ---------------------------------------------------
extern "C" void kernel_launch(void* const* d_in, const int* in_sizes, int n_in,
                              void* d_out, int out_size, void* d_ws, size_t ws_size,
                              hipStream_t stream) {
  (void)in_sizes; (void)n_in; (void)out_size; (void)ws_size;
  const float* xf  = (const float*)d_in[0];
  const float* xh1 = (const float*)d_in[1];
  const float* xh2 = (const float*)d_in[2];
  float* ws = (float*)d_ws;

  fold_kernel<<<3, 256, 0, stream>>>(
      (const float*)d_in[3],  (const float*)d_in[4],  (const float*)d_in[5],
      (const float*)d_in[6],  (const float*)d_in[7],
      (const float*)d_in[8],  (const float*)d_in[9],  (const float*)d_in[10],
      (const float*)d_in[11], (const float*)d_in[12],
      (const float*)d_in[13], (const float*)d_in[14], (const float*)d_in[15],
      (const float*)d_in[16], (const float*)d_in[17],
      (const float*)d_in[18], (const float*)d_in[19], (const float*)d_in[20],
      (const float*)d_in[21], (const float*)d_in[22],
      ws);

  fused_kernel<<<2048, 256, 0, stream>>>(xf, xh1, xh2, ws, (float*)d_out);
}